// LIFLayer_2894807957745
// MI455X (gfx1250) — compile-verified
//
#include <hip/hip_runtime.h>

// LIF layer: syn_new = S*syn + w*spike_row (rank-1), colsum = syn_new.sum(rows),
// v_new = V*v + colsum - (V*v >= 1), spikes = (V*v >= 1).
// Memory-bound (~805 MB @ 23.3 TB/s ~= 35us). WMMA f32 16x16x4 with A=ones does
// the column reduction inside the streaming pass.

#define N_IN  8192
#define N_OUT 8192
#define ROWS_PER_BLOCK 128
#define COLS_PER_WAVE  64
#define WAVES_PER_BLOCK 8
#define COLS_PER_BLOCK (COLS_PER_WAVE * WAVES_PER_BLOCK)   // 512

typedef __attribute__((ext_vector_type(2))) float v2f;
typedef __attribute__((ext_vector_type(4))) float v4f;
typedef __attribute__((ext_vector_type(8))) float v8f;

__constant__ constexpr float V_DECAY = 0.99004983374916811f;  // exp(-1/100)
__constant__ constexpr float S_DECAY = 0.98019867330675525f;  // exp(-1/50)

__device__ __forceinline__ v4f ld4(const float* p) { return *(const v4f*)p; }

__global__ void __launch_bounds__(256)
lif_zero_ws(float* __restrict__ colsum) {
  colsum[blockIdx.x * 256 + threadIdx.x] = 0.0f;
}

__global__ void __launch_bounds__(256)
lif_main(const float* __restrict__ spk,     // [N_IN]
         const float* __restrict__ syn,     // [N_IN, N_OUT]
         const float* __restrict__ wgt,     // [N_IN, N_OUT]
         float* __restrict__ syn_out,       // [N_IN, N_OUT]
         float* __restrict__ colsum) {      // [N_OUT] f32 atomics
  __shared__ float sSpk[ROWS_PER_BLOCK];
  const int tid  = threadIdx.x;
  const int wv   = tid >> 5;
  const int lane = tid & 31;
  const int n    = lane & 15;   // B-matrix column position
  const int k    = lane >> 4;   // half-wave selects K rows
  const int c0   = blockIdx.x * COLS_PER_BLOCK + wv * COLS_PER_WAVE;
  const int r0   = blockIdx.y * ROWS_PER_BLOCK;

  if (tid < ROWS_PER_BLOCK) sSpk[tid] = spk[r0 + tid];
  __syncthreads();

  const int colbase = c0 + 4 * n;   // this lane's 4 actual columns
  v8f acc0 = {}, acc1 = {}, acc2 = {}, acc3 = {};
  const v2f ones = {1.0f, 1.0f};

  for (int rr = 0; rr < ROWS_PER_BLOCK; rr += 4) {
    const int ra = rr + k;        // local rows: k=0 -> {rr, rr+2}, k=1 -> {rr+1, rr+3}
    const int rb = rr + 2 + k;
    const int ia = (r0 + ra) * N_OUT + colbase;
    const int ib = (r0 + rb) * N_OUT + colbase;

    v4f sA = ld4(syn + ia), sB = ld4(syn + ib);   // global_load_b128
    v4f wA = ld4(wgt + ia), wB = ld4(wgt + ib);
    const float fa = sSpk[ra], fb = sSpk[rb];

    v4f oA, oB;
#pragma unroll
    for (int q = 0; q < 4; ++q) {
      oA[q] = sA[q] * S_DECAY + wA[q] * fa;
      oB[q] = sB[q] * S_DECAY + wB[q] * fb;
    }
    *(v4f*)(syn_out + ia) = oA;                    // global_store_b128
    *(v4f*)(syn_out + ib) = oB;

    // Each WMMA q reduces 4 rows of column (colbase+q) at B position n.
    // A = ones => D[m][n] = C[m][n] + sum_k B[k][n]; K-slot permutation irrelevant.
    v2f b0 = {oA[0], oB[0]};
    v2f b1 = {oA[1], oB[1]};
    v2f b2 = {oA[2], oB[2]};
    v2f b3 = {oA[3], oB[3]};
    acc0 = __builtin_amdgcn_wmma_f32_16x16x4_f32(false, ones, false, b0, (short)0, acc0, false, false);
    acc1 = __builtin_amdgcn_wmma_f32_16x16x4_f32(false, ones, false, b1, (short)0, acc1, false, false);
    acc2 = __builtin_amdgcn_wmma_f32_16x16x4_f32(false, ones, false, b2, (short)0, acc2, false, false);
    acc3 = __builtin_amdgcn_wmma_f32_16x16x4_f32(false, ones, false, b3, (short)0, acc3, false, false);
  }

  // C/D layout: VGPR0, lanes 0..15 = (M=0, N=lane); rows replicated, take M=0.
  if (lane < 16) {
    atomicAdd(&colsum[colbase + 0], acc0[0]);
    atomicAdd(&colsum[colbase + 1], acc1[0]);
    atomicAdd(&colsum[colbase + 2], acc2[0]);
    atomicAdd(&colsum[colbase + 3], acc3[0]);
  }
}

__global__ void __launch_bounds__(256)
lif_final(const float* __restrict__ volt,     // [N_OUT]
          const float* __restrict__ colsum,   // [N_OUT]
          float* __restrict__ out_sp,         // [N_OUT] spikes as 0/1 float
          float* __restrict__ out_v) {        // [N_OUT]
  const int j = blockIdx.x * 256 + threadIdx.x;
  const float v = volt[j] * V_DECAY;
  const float s = (v >= 1.0f) ? 1.0f : 0.0f;
  out_sp[j] = s;
  out_v[j]  = v + colsum[j] - s;
}

extern "C" void kernel_launch(void* const* d_in, const int* in_sizes, int n_in,
                              void* d_out, int out_size, void* d_ws, size_t ws_size,
                              hipStream_t stream) {
  const float* spk  = (const float*)d_in[0];   // input_spikes [8192]
  const float* volt = (const float*)d_in[1];   // voltages    [1,8192]
  const float* syn  = (const float*)d_in[2];   // synapses    [8192,8192]
  const float* wgt  = (const float*)d_in[3];   // weights     [8192,8192]

  float* out      = (float*)d_out;
  float* out_sp   = out;                       // spikes (8192)
  float* out_v    = out + N_OUT;               // v_new  (8192)
  float* syn_out  = out + 2 * N_OUT;           // syn    (8192*8192)
  float* colsum   = (float*)d_ws;              // 8192 f32 scratch

  lif_zero_ws<<<N_OUT / 256, 256, 0, stream>>>(colsum);

  dim3 grid(N_OUT / COLS_PER_BLOCK, N_IN / ROWS_PER_BLOCK);   // 16 x 64
  lif_main<<<grid, 256, 0, stream>>>(spk, syn, wgt, syn_out, colsum);

  lif_final<<<N_OUT / 256, 256, 0, stream>>>(volt, colsum, out_sp, out_v);
}